// tokenselector_79362405695530
// MI455X (gfx1250) — compile-verified
//
#include <hip/hip_runtime.h>
#include <hip/hip_bf16.h>
#include <math.h>

// Problem constants
#define KB    16      // batch
#define KDIM  768     // channels
#define KHW   1024    // 32*32 pixels
#define KNT   32      // tokens
#define CHUNK 32      // channels per K-chunk staged in LDS
#define ROWS  2       // image rows per block tile (P = 64 pixels)
#define TP    64      // pixels per tile
#define NC    (KDIM / CHUNK)               // 24 chunks
#define FELEM (CHUNK * (ROWS + 2) * 34)    // 4352 halo elements per chunk
#define FPT   (FELEM / 256)                // 17 async copies per thread per chunk

// d_out layout: result[16*32*768] | p[16*32*768] | mask[16*32*1024]
#define P_OFF  (KB * KNT * KDIM)           // 393216
#define M_OFF  (2 * KB * KNT * KDIM)       // 786432

typedef __attribute__((ext_vector_type(2))) float v2f;
typedef __attribute__((ext_vector_type(8))) float v8f;

#if __has_builtin(__builtin_amdgcn_global_load_async_to_lds_b32) && \
    __has_builtin(__builtin_amdgcn_s_wait_asynccnt)
#define USE_ASYNC 1
typedef __attribute__((address_space(1))) int* gas_i32;  // global AS, int*
typedef __attribute__((address_space(3))) int* las_i32;  // LDS AS, int*
#else
#define USE_ASYNC 0
#endif

// -------------------------------------------------------------------------
// Kernel 1: fused BN -> depthwise 3x3 -> hardswish -> pointwise GEMM (WMMA)
// Raw x is staged to LDS with ASYNC global->LDS copies, double-buffered so
// chunk c+1's HBM traffic overlaps chunk c's dwconv + WMMA. BN is folded
// into the dwconv via  scale * (sum x*w) + shift * (sum_valid w) + dw_bias.
// Writes s (pre-mask scores) into the mask region of d_out.
// grid = KB * 16 (16 two-row tiles per batch), block = 256 (8 waves)
// -------------------------------------------------------------------------
__global__ __launch_bounds__(256) void fused_score_kernel(
    const float* __restrict__ x,
    const float* __restrict__ bn_w, const float* __restrict__ bn_b,
    const float* __restrict__ bn_m, const float* __restrict__ bn_v,
    const float* __restrict__ dw_w, const float* __restrict__ dw_b,
    const float* __restrict__ pw_w, const float* __restrict__ pw_b,
    float* __restrict__ s_out)
{
    __shared__ float f_lds[2][CHUNK][ROWS + 2][34]; // raw x tiles + halo (double buffered)
    __shared__ float h_lds[CHUNK][72];              // hardswish(dwconv), padded stride
    __shared__ float w_lds[KNT][33];                // pw weights chunk, padded stride
    __shared__ float dww_lds[CHUNK][9];             // dw weights chunk
    __shared__ float sc_lds[CHUNK], sh_lds[CHUNK], db_lds[CHUNK];
    __shared__ float dummy_lds[16];                 // sink for OOB async copies

    const int tid  = threadIdx.x;
    const int blk  = blockIdx.x;
    const int b    = blk >> 4;          // batch
    const int tile = blk & 15;          // row tile
    const int row0 = tile * ROWS;
    const long xbase = (long)b * KDIM * KHW;

    const int wv   = tid >> 5;          // wave 0..7
    const int lane = tid & 31;
    const int lm   = lane & 15;
    const int kh   = (lane >> 4) * 2;   // K-half select: 0 or 2
    const int mbase = (wv >> 2) * 16;   // token tile base (0 or 16)
    const int p0    = (wv & 3) * 16;    // pixel tile base

    v8f cacc = {};                       // one 16x16 f32 accumulator per wave

    // ---- prologue: zero the OOB halo cells of both buffers (async copies
    // never touch them), then kick off chunk 0's async staging ----
    for (int i = tid; i < FELEM; i += 256) {
        int c   = i / ((ROWS + 2) * 34);
        int rem = i % ((ROWS + 2) * 34);
        int r   = rem / 34;
        int cc  = rem % 34;
        int gr  = row0 + r - 1, gc = cc - 1;
        if (!(gr >= 0 && gr < 32 && gc >= 0 && gc < 32)) {
            f_lds[0][c][r][cc] = 0.0f;
            f_lds[1][c][r][cc] = 0.0f;
        }
    }

    // stage chunk `c0` of raw x into buffer `buf` (17 copies per thread, uniform)
    auto stage = [&](int c0, int buf) {
        #pragma unroll
        for (int k = 0; k < FPT; ++k) {
            int i   = tid + k * 256;
            int c   = i / ((ROWS + 2) * 34);
            int rem = i % ((ROWS + 2) * 34);
            int r   = rem / 34;
            int cc  = rem % 34;
            int gr  = row0 + r - 1, gc = cc - 1;
            bool ib = (gr >= 0 && gr < 32 && gc >= 0 && gc < 32);
            const float* gsrc = ib ? (x + xbase + (long)(c0 + c) * KHW + gr * 32 + gc)
                                   : x;                       // dummy but valid
            float* ldst = ib ? &f_lds[buf][c][r][cc] : &dummy_lds[tid & 15];
#if USE_ASYNC
            __builtin_amdgcn_global_load_async_to_lds_b32(
                (gas_i32)(size_t)gsrc, (las_i32)ldst, 0, 0);
#else
            if (ib) *ldst = *gsrc;
#endif
        }
    };

    stage(0, 0);

    for (int cidx = 0; cidx < NC; ++cidx) {
        const int c0  = cidx * CHUNK;
        const int buf = cidx & 1;

        // ---- issue next chunk's async copies, then wait for this chunk ----
        if (cidx + 1 < NC) {
            stage(c0 + CHUNK, buf ^ 1);
#if USE_ASYNC
            __builtin_amdgcn_s_wait_asynccnt(FPT);  // newest FPT = next chunk
#endif
        } else {
#if USE_ASYNC
            __builtin_amdgcn_s_wait_asynccnt(0);
#endif
        }
        __syncthreads();

        // ---- stage per-channel constants and weights (small, L2-hot) ----
        if (tid < CHUNK) {
            int c = c0 + tid;
            float sc = bn_w[c] * rsqrtf(bn_v[c] + 1e-5f);
            sc_lds[tid] = sc;
            sh_lds[tid] = bn_b[c] - bn_m[c] * sc;
            db_lds[tid] = dw_b[c];
        }
        for (int i = tid; i < CHUNK * 9; i += 256)
            dww_lds[i / 9][i % 9] = dw_w[(c0 + i / 9) * 9 + (i % 9)];
        for (int i = tid; i < KNT * CHUNK; i += 256) {
            int t = i / CHUNK, c = i % CHUNK;
            w_lds[t][c] = pw_w[t * KDIM + c0 + c];
        }
        __syncthreads();

        // ---- depthwise 3x3 on raw x with BN folded in, + bias, hardswish ----
        for (int i = tid; i < CHUNK * TP; i += 256) {
            int c   = i / TP;
            int p   = i % TP;
            int r   = p >> 5;
            int col = p & 31;
            float acc = 0.0f, ws = 0.0f;
            #pragma unroll
            for (int dr = 0; dr < 3; ++dr) {
                bool rok = (unsigned)(row0 + r - 1 + dr) < 32u;
                #pragma unroll
                for (int dc = 0; dc < 3; ++dc) {
                    float w = dww_lds[c][dr * 3 + dc];
                    acc += f_lds[buf][c][r + dr][col + dc] * w;
                    if (rok && (unsigned)(col - 1 + dc) < 32u) ws += w;
                }
            }
            float pre = sc_lds[c] * acc + sh_lds[c] * ws + db_lds[c];
            float t = pre + 3.0f;
            t = fminf(fmaxf(t, 0.0f), 6.0f);
            h_lds[c][p] = pre * t * (1.0f / 6.0f);
        }
        __syncthreads();

        // ---- pointwise GEMM chunk via V_WMMA_F32_16X16X4_F32 ----
        // A (16x4): lanes 0-15 -> M=lm, K=kh..kh+1 ; lanes 16-31 -> K=2..3
        // B (4x16): lanes 0-15 -> K=kh..kh+1, N=lm ; lanes 16-31 -> K=2..3
        #pragma unroll
        for (int ks = 0; ks < CHUNK / 4; ++ks) {
            int ck = ks * 4;
            v2f a, bb;
            a.x  = w_lds[mbase + lm][ck + kh];
            a.y  = w_lds[mbase + lm][ck + kh + 1];
            bb.x = h_lds[ck + kh][p0 + lm];
            bb.y = h_lds[ck + kh + 1][p0 + lm];
            cacc = __builtin_amdgcn_wmma_f32_16x16x4_f32(
                false, a, false, bb, (short)0, cacc, false, false);
        }
        __syncthreads();  // protect h_lds/w_lds before next chunk overwrites
    }

    // ---- store s tile (+ pw_bias) into mask region ----
    // C/D layout: VGPR e, lanes 0-15 -> M=e, N=lane ; lanes 16-31 -> M=e+8
    const int pix  = row0 * 32 + p0 + lm;
    const int mrow = (lane >> 4) * 8;
    #pragma unroll
    for (int e = 0; e < 8; ++e) {
        int tok = mbase + mrow + e;
        s_out[((long)b * KNT + tok) * KHW + pix] = cacc[e] + pw_b[tok];
    }
}

// -------------------------------------------------------------------------
// Kernel 2: per-(batch, token) spatial argmax -> mask (in place over s),
// then gather-sum of BN(x) and pos at matching positions.
// grid = KB*KNT = 512, block = 256
// -------------------------------------------------------------------------
__global__ __launch_bounds__(256) void argmax_gather_kernel(
    const float* __restrict__ x, const float* __restrict__ pos,
    const float* __restrict__ bn_w, const float* __restrict__ bn_b,
    const float* __restrict__ bn_m, const float* __restrict__ bn_v,
    float* __restrict__ out)
{
    const int tid = threadIdx.x;
    const int b   = blockIdx.x >> 5;
    const int t   = blockIdx.x & 31;
    float* s = out + M_OFF + ((long)b * KNT + t) * KHW;

    // phase 1: block-wide max over 1024 scores
    float v[4];
    float local = -INFINITY;
    #pragma unroll
    for (int j = 0; j < 4; ++j) {
        v[j] = s[tid + j * 256];
        local = fmaxf(local, v[j]);
    }
    #pragma unroll
    for (int off = 16; off > 0; off >>= 1)
        local = fmaxf(local, __shfl_xor(local, off, 32));

    __shared__ float wmax[8];
    __shared__ int npos;
    __shared__ int plist[64];
    if ((tid & 31) == 0) wmax[tid >> 5] = local;
    if (tid == 0) npos = 0;
    __syncthreads();
    float m = wmax[0];
    #pragma unroll
    for (int i = 1; i < 8; ++i) m = fmaxf(m, wmax[i]);

    // phase 2: write mask in place, collect matching positions (ties summed)
    #pragma unroll
    for (int j = 0; j < 4; ++j) {
        int p = tid + j * 256;
        float mk = (v[j] == m) ? 1.0f : 0.0f;
        s[p] = mk;
        if (mk != 0.0f) {
            int idx = atomicAdd(&npos, 1);
            if (idx < 64) plist[idx] = p;
        }
    }
    __syncthreads();
    int n = npos < 64 ? npos : 64;

    // phase 3: gather-sum over channels (pos is only touched at argmax cols)
    const long xb = ((long)b * KDIM) * KHW;
    for (int ch = tid; ch < KDIM; ch += 256) {
        float sc = bn_w[ch] * rsqrtf(bn_v[ch] + 1e-5f);
        float sh = bn_b[ch] - bn_m[ch] * sc;
        const float* xp = x   + xb + (long)ch * KHW;
        const float* pp = pos + xb + (long)ch * KHW;
        float ar = 0.0f, ap = 0.0f;
        for (int i = 0; i < n; ++i) {
            int p = plist[i];
            ar += xp[p] * sc + sh;
            ap += pp[p];
        }
        long o = ((long)b * KNT + t) * KDIM + ch;
        out[o]         = ar;
        out[P_OFF + o] = ap;
    }
}

extern "C" void kernel_launch(void* const* d_in, const int* in_sizes, int n_in,
                              void* d_out, int out_size, void* d_ws, size_t ws_size,
                              hipStream_t stream) {
    const float* x    = (const float*)d_in[0];
    const float* pos  = (const float*)d_in[1];
    const float* bn_w = (const float*)d_in[2];
    const float* bn_b = (const float*)d_in[3];
    const float* bn_m = (const float*)d_in[4];
    const float* bn_v = (const float*)d_in[5];
    const float* dw_w = (const float*)d_in[6];
    const float* dw_b = (const float*)d_in[7];
    const float* pw_w = (const float*)d_in[8];
    const float* pw_b = (const float*)d_in[9];
    float* out = (float*)d_out;

    // Kernel 1: 16 batches x 16 row-tiles; writes s into mask region of out.
    fused_score_kernel<<<dim3(KB * 16), dim3(256), 0, stream>>>(
        x, bn_w, bn_b, bn_m, bn_v, dw_w, dw_b, pw_w, pw_b, out + M_OFF);

    // Kernel 2: one block per (batch, token); same stream => ordered.
    argmax_gather_kernel<<<dim3(KB * KNT), dim3(256), 0, stream>>>(
        x, pos, bn_w, bn_b, bn_m, bn_v, out);
}